// MPNNLayer_4140348474008
// MI455X (gfx1250) — compile-verified
//
#include <hip/hip_runtime.h>

typedef __attribute__((ext_vector_type(2))) float v2f;
typedef __attribute__((ext_vector_type(4))) float v4f;
typedef __attribute__((ext_vector_type(8))) float v8f;

// Problem sizes (fixed by reference): B=16, N=128, D=128
// Row count for all panel GEMMs: M = B*N = 2048, K = 128, Ncols = 128.

// ---------------------------------------------------------------------------
// Kernel A: streaming pass over e (134 MB, HBM-bound).
//   es[p][b,j,d] = sum_{i in half p} adj[b,i,j] * e[b,i,j,d]
//   ah[p][b,j,d] = sum_{i in half p} adj[b,i,j] * h[b,i,d]
//   deg[p][b,j]  = sum_{i in half p} adj[b,i,j]
// Split-i: two waves per (b,j) (4096 waves total) for HBM latency hiding.
// Lane l covers d = 4l..4l+3 (float4): each wave reads contiguous 512B rows
// of e, non-temporally (zero reuse; keep L2 for h/adj/weights).
// ---------------------------------------------------------------------------
__global__ void __launch_bounds__(256) reduce_e_kernel(
    const float* __restrict__ e, const float* __restrict__ adj,
    const float* __restrict__ h, float* __restrict__ es,
    float* __restrict__ ah, float* __restrict__ deg) {
  const int lane = threadIdx.x & 31;
  const int wave = threadIdx.x >> 5;
  const int w  = blockIdx.x * 8 + wave;   // 0..4095
  const int bj = w >> 1;                  // 0..2047
  const int p  = w & 1;                   // i-half: [64p, 64p+64)
  const int b = bj >> 7;
  const int j = bj & 127;

  // Preload this wave's half of the adj column adj[b, :, j]: i = 64p+32c+lane.
  float acol[2];
#pragma unroll
  for (int c = 0; c < 2; ++c)
    acol[c] = adj[((size_t)b * 128 + p * 64 + c * 32 + lane) * 128 + j];

  const v4f* e4 = (const v4f*)e;
  const v4f* h4 = (const v4f*)h;
  // e[b,i,j,4*lane]: f4 index = ((b*128+i)*128 + j)*32 + lane
  const v4f* ebase = e4 + ((size_t)b * 128 * 128 + j) * 32 + lane;
  // h[b,i,4*lane]:  f4 index = (b*128+i)*32 + lane
  const v4f* hbase = h4 + (size_t)b * 128 * 32 + lane;

  v4f accE = {0.f, 0.f, 0.f, 0.f};
  v4f accH = {0.f, 0.f, 0.f, 0.f};
#pragma unroll
  for (int c = 0; c < 2; ++c) {
#pragma unroll 8
    for (int ii = 0; ii < 32; ++ii) {
      const int i = p * 64 + c * 32 + ii;
      const float a = __shfl(acol[c], ii, 32);
      v4f ev = __builtin_nontemporal_load(ebase + (size_t)i * (128 * 32));
      v4f hv = hbase[(size_t)i * 32];
      accE.x += a * ev.x; accE.y += a * ev.y;
      accE.z += a * ev.z; accE.w += a * ev.w;
      accH.x += a * hv.x; accH.y += a * hv.y;
      accH.z += a * hv.z; accH.w += a * hv.w;
    }
  }
  const size_t part = (size_t)p * 2048 * 32;  // f4 units per partial buffer
  ((v4f*)es)[part + (size_t)bj * 32 + lane] = accE;
  ((v4f*)ah)[part + (size_t)bj * 32 + lane] = accH;

  float s = acol[0] + acol[1];
#pragma unroll
  for (int off = 16; off > 0; off >>= 1) s += __shfl_xor(s, off, 32);
  if (lane == 0) deg[p * 2048 + bj] = s;
}

// ---------------------------------------------------------------------------
// Wave-level fp32 WMMA GEMM tile: C(16x16) += A(16x128) * B(16x128)^T.
// Both operands row-major with the K=128 contraction on the fast axis (NT).
// ISA 16x4 fragment layout: lanes 0-15 hold rows 0-15 with K={k,k+1},
// lanes 16-31 hold the same rows with K={k+2,k+3} -> one float2 per lane.
// ---------------------------------------------------------------------------
__device__ __forceinline__ v8f wmma_nt_k128(const float* __restrict__ A, int lda,
                                            const float* __restrict__ Bm, int ldb,
                                            int tm, int tn, v8f acc, int lane) {
  const int hf = lane >> 4;
  const int r  = lane & 15;
  const float* arow = A  + (size_t)(tm + r) * lda + 2 * hf;
  const float* brow = Bm + (size_t)(tn + r) * ldb + 2 * hf;
#pragma unroll 8
  for (int kk = 0; kk < 128; kk += 4) {
    v2f av = *(const v2f*)(arow + kk);
    v2f bv = *(const v2f*)(brow + kk);
    acc = __builtin_amdgcn_wmma_f32_16x16x4_f32(
        /*neg_a=*/false, av, /*neg_b=*/false, bv,
        /*c_mod=*/(short)0, acc, /*reuse_a=*/false, /*reuse_b=*/false);
  }
  return acc;
}

// ---------------------------------------------------------------------------
// Kernel B: msum[m,k] = (ah0+ah1)[m,:]@W1[k,:] + (es0+es1)[m,:]@W3[k,:]
//                       + (deg0+deg1)[m] * ( h[m,:]@W2[k,:] + bm[k] )
// Partial sums folded in as extra WMMA accumulation chains (GEMM distributes
// over the sum). Wm is (128,384) row-major: W1=Wm+0, W2=Wm+128, W3=Wm+256.
// One wave per 16x16 tile; 2048x128 output -> 1024 tiles -> 128 blocks.
// ---------------------------------------------------------------------------
__global__ void __launch_bounds__(256) msum_kernel(
    const float* __restrict__ h, const float* __restrict__ ah,
    const float* __restrict__ es, const float* __restrict__ Wm,
    const float* __restrict__ bm, const float* __restrict__ deg,
    float* __restrict__ msum) {
  const int lane = threadIdx.x & 31;
  const int wave = threadIdx.x >> 5;
  const int tile = blockIdx.x * 8 + wave;  // 0..1023
  const int tM = (tile >> 3) << 4;
  const int tN = (tile & 7) << 4;
  const size_t part = (size_t)2048 * 128;  // floats per partial buffer

  v8f acc1 = {};
  acc1 = wmma_nt_k128(ah,        128, Wm + 0,   384, tM, tN, acc1, lane);
  acc1 = wmma_nt_k128(ah + part, 128, Wm + 0,   384, tM, tN, acc1, lane);
  acc1 = wmma_nt_k128(es,        128, Wm + 256, 384, tM, tN, acc1, lane);
  acc1 = wmma_nt_k128(es + part, 128, Wm + 256, 384, tM, tN, acc1, lane);
  v8f acc2 = {};
  acc2 = wmma_nt_k128(h,         128, Wm + 128, 384, tM, tN, acc2, lane);

  const int hf = lane >> 4;
  const int r  = lane & 15;
  const float bmv = bm[tN + r];
#pragma unroll
  for (int v = 0; v < 8; ++v) {
    const int row = tM + v + 8 * hf;       // C layout: VGPR v, half hf -> M = v + 8*hf
    const float d = deg[row] + deg[2048 + row];
    msum[(size_t)row * 128 + tN + r] = acc1[v] + d * (acc2[v] + bmv);
  }
}

// ---------------------------------------------------------------------------
// Kernel C: out[m,k] = h[m,:]@Wu[k,0:128] + msum[m,:]@Wu[k,128:256] + bu[k]
// Wu is (128,256) row-major, ldb=256.
// ---------------------------------------------------------------------------
__global__ void __launch_bounds__(256) out_kernel(
    const float* __restrict__ h, const float* __restrict__ msum,
    const float* __restrict__ Wu, const float* __restrict__ bu,
    float* __restrict__ out) {
  const int lane = threadIdx.x & 31;
  const int wave = threadIdx.x >> 5;
  const int tile = blockIdx.x * 8 + wave;  // 0..1023
  const int tM = (tile >> 3) << 4;
  const int tN = (tile & 7) << 4;

  v8f acc = {};
  acc = wmma_nt_k128(h,    128, Wu + 0,   256, tM, tN, acc, lane);
  acc = wmma_nt_k128(msum, 128, Wu + 128, 256, tM, tN, acc, lane);

  const int hf = lane >> 4;
  const int r  = lane & 15;
  const float buv = bu[tN + r];
#pragma unroll
  for (int v = 0; v < 8; ++v) {
    const int row = tM + v + 8 * hf;
    out[(size_t)row * 128 + tN + r] = acc[v] + buv;
  }
}

extern "C" void kernel_launch(void* const* d_in, const int* in_sizes, int n_in,
                              void* d_out, int out_size, void* d_ws, size_t ws_size,
                              hipStream_t stream) {
  const float* h   = (const float*)d_in[0];  // (16,128,128)
  const float* adj = (const float*)d_in[1];  // (16,128,128)
  const float* e   = (const float*)d_in[2];  // (16,128,128,128)
  const float* Wm  = (const float*)d_in[3];  // (128,384)
  const float* bm  = (const float*)d_in[4];  // (128,)
  const float* Wu  = (const float*)d_in[5];  // (128,256)
  const float* bu  = (const float*)d_in[6];  // (128,)
  float* out = (float*)d_out;                // (16,128,128)

  // Workspace carve-up (~5.2 MB): es[2 parts], ah[2 parts], msum, deg[2 parts]
  const size_t P = (size_t)2048 * 128;
  float* es   = (float*)d_ws;        // 2*P
  float* ah   = es   + 2 * P;        // 2*P
  float* msum = ah   + 2 * P;        // P
  float* deg  = msum + P;            // 2*2048

  reduce_e_kernel<<<512, 256, 0, stream>>>(e, adj, h, es, ah, deg);
  msum_kernel<<<128, 256, 0, stream>>>(h, ah, es, Wm, bm, deg, msum);
  out_kernel<<<128, 256, 0, stream>>>(h, msum, Wu, bu, out);
}